// HeteroRGCN_26310969655542
// MI455X (gfx1250) — compile-verified
//
#include <hip/hip_runtime.h>
#include <math.h>

#define NNODES 20000
#define NEDGES 500000
#define INF_F  128
#define HIDF   128
#define OUTF   64
#define NEG_SLOPE 0.01f

typedef __attribute__((ext_vector_type(2))) float v2f;
typedef __attribute__((ext_vector_type(8))) float v8f;

// ---------------------------------------------------------------- utilities
__global__ void fill_kernel(float* __restrict__ p, float v, int n) {
    int i = blockIdx.x * blockDim.x + threadIdx.x;
    if (i < n) p[i] = v;
}

__global__ void leaky_relu_kernel(float* __restrict__ p, int n) {
    int i = blockIdx.x * blockDim.x + threadIdx.x;
    if (i < n) {
        float v = p[i];
        p[i] = v > 0.f ? v : NEG_SLOPE * v;
    }
}

__device__ __forceinline__ void atomicMaxFloat(float* addr, float val) {
    // Monotone bit-pattern trick: int-max for >=0, uint-min for <0.
    if (val >= 0.f)
        atomicMax((int*)addr, __float_as_int(val));
    else
        atomicMin((unsigned int*)addr, __float_as_uint(val));
}

// ------------------------------------------------- GEMM: Wh = X * W^T + b
// X: [nrows, K] row-major; W: [outf, K] row-major (torch Linear layout).
// One wave computes one 16x16 f32 tile via V_WMMA_F32_16X16X4_F32, K step 4.
__global__ void gemm_wmma_f32(const float* __restrict__ X,
                              const float* __restrict__ W,
                              const float* __restrict__ bias,
                              float* __restrict__ Wh,
                              int nrows, int K, int outf) {
    int wave = (blockIdx.x * blockDim.x + threadIdx.x) >> 5;
    int lane = threadIdx.x & 31;
    int tilesN = outf >> 4;
    int tilesM = nrows >> 4;            // NNODES % 16 == 0
    if (wave >= tilesM * tilesN) return;  // uniform across the wave -> EXEC all-1s below
    int tm = wave / tilesN;
    int tn = wave % tilesN;

    int half = lane >> 4;               // 0: lanes 0-15, 1: lanes 16-31
    int row  = tm * 16 + (lane & 15);   // A-matrix M index for this lane
    int col  = tn * 16 + (lane & 15);   // B/C-matrix N index for this lane

    const float* xrow = X + (size_t)row * K;
    const float* wrow = W + (size_t)col * K;   // B[k][col] = W[col][k]

    v8f acc = {};
    #pragma unroll 8
    for (int k0 = 0; k0 < K; k0 += 4) {
        int ka = k0 + 2 * half;
        v2f a, b;
        a.x = xrow[ka];     // A VGPR0: K = 2*half
        a.y = xrow[ka + 1]; // A VGPR1: K = 2*half + 1
        b.x = wrow[ka];     // B VGPR0: K = 2*half
        b.y = wrow[ka + 1]; // B VGPR1: K = 2*half + 1
        acc = __builtin_amdgcn_wmma_f32_16x16x4_f32(
                  false, a, false, b, (short)0, acc, false, false);
    }

    float bv = bias[col];
    #pragma unroll
    for (int v = 0; v < 8; ++v) {
        int m = tm * 16 + v + half * 8;     // C layout: VGPR v -> M = v (+8 for hi half)
        Wh[(size_t)m * outf + col] = acc[v] + bv;
    }
}

// ---------------------------------------- al = Wh @ a[:outf], ar = Wh @ a[outf:]
__global__ void attn_proj_kernel(const float* __restrict__ Wh,
                                 const float* __restrict__ avec,
                                 float* __restrict__ al, float* __restrict__ ar,
                                 int n, int outf) {
    int i = blockIdx.x * blockDim.x + threadIdx.x;
    if (i >= n) return;
    const float4* r  = (const float4*)(Wh + (size_t)i * outf);
    const float4* gl = (const float4*)avec;
    const float4* gr = (const float4*)(avec + outf);
    float sl = 0.f, sr = 0.f;
    int q = outf >> 2;
    for (int f = 0; f < q; ++f) {
        float4 v = r[f];
        float4 cl = gl[f], cr = gr[f];
        sl += v.x * cl.x + v.y * cl.y + v.z * cl.z + v.w * cl.w;
        sr += v.x * cr.x + v.y * cr.y + v.z * cr.z + v.w * cr.w;
    }
    al[i] = sl;
    ar[i] = sr;
}

// ------------------------- edge pass 1: e = leakyrelu(al[src]+ar[dst]); seg-max
__global__ void edge_pass1_kernel(const int* __restrict__ src, const int* __restrict__ dst,
                                  const float* __restrict__ al, const float* __restrict__ ar,
                                  float* __restrict__ ebuf, float* __restrict__ m, int ne) {
    int i = blockIdx.x * blockDim.x + threadIdx.x;
    if (i >= ne) return;
    float e = al[src[i]] + ar[dst[i]];
    e = e > 0.f ? e : NEG_SLOPE * e;
    ebuf[i] = e;
    atomicMaxFloat(&m[dst[i]], e);
}

// ------------------------- edge pass 2: ex = exp(e - m[dst]); seg-sum denom
__global__ void edge_pass2_kernel(const int* __restrict__ dst,
                                  const float* __restrict__ ebuf,
                                  const float* __restrict__ m,
                                  float* __restrict__ exbuf, float* __restrict__ denom,
                                  int ne) {
    int i = blockIdx.x * blockDim.x + threadIdx.x;
    if (i >= ne) return;
    int d = dst[i];
    float ex = __expf(ebuf[i] - m[d]);
    exbuf[i] = ex;
    atomicAdd(&denom[d], ex);
}

// ---- edge pass 3: h[dst] += (ex/denom[dst]) * Wh[src]   (wave per edge)
// Each lane owns a CONTIGUOUS feature chunk so the row gather is a single
// global_load_b128 (outf=128) / b64 (outf=64) per lane, fully coalesced.
__global__ void edge_aggregate_kernel(const int* __restrict__ src, const int* __restrict__ dst,
                                      const float* __restrict__ exbuf,
                                      const float* __restrict__ denom,
                                      const float* __restrict__ Wh,
                                      float* __restrict__ h,
                                      int ne, int outf) {
    int wave = (blockIdx.x * blockDim.x + threadIdx.x) >> 5;
    int lane = threadIdx.x & 31;
    if (wave >= ne) return;
    int s = src[wave];
    int d = dst[wave];
    float alpha = exbuf[wave] / denom[d];
    const float* wr = Wh + (size_t)s * outf;
    float* hr = h + (size_t)d * outf;
    if (outf == 128) {
        float4 v = ((const float4*)wr)[lane];           // 512B coalesced per wave
        int f = lane * 4;
        atomicAdd(&hr[f + 0], alpha * v.x);
        atomicAdd(&hr[f + 1], alpha * v.y);
        atomicAdd(&hr[f + 2], alpha * v.z);
        atomicAdd(&hr[f + 3], alpha * v.w);
    } else {                                            // outf == 64
        float2 v = ((const float2*)wr)[lane];           // 256B coalesced per wave
        int f = lane * 2;
        atomicAdd(&hr[f + 0], alpha * v.x);
        atomicAdd(&hr[f + 1], alpha * v.y);
    }
}

// ---------------------------------------------------------------- driver
static void run_layer(const float* Xin,
                      const float* W[2], const float* b[2], const float* av[2],
                      const int* src[2], const int* dst[2],
                      float* hout, int K, int outf,
                      float* Wh, float* al, float* ar, float* m, float* dn,
                      float* ebuf, float* exbuf, hipStream_t stream) {
    const int NE = NEDGES;
    hipMemsetAsync(hout, 0, (size_t)NNODES * outf * sizeof(float), stream);
    for (int et = 0; et < 2; ++et) {
        // GEMM + bias
        int tiles = (NNODES / 16) * (outf / 16);
        int waves_per_block = 4;                       // 128 threads
        int grid = (tiles + waves_per_block - 1) / waves_per_block;
        gemm_wmma_f32<<<grid, 128, 0, stream>>>(Xin, W[et], b[et], Wh, NNODES, K, outf);
        // attention projections
        attn_proj_kernel<<<(NNODES + 255) / 256, 256, 0, stream>>>(Wh, av[et], al, ar,
                                                                   NNODES, outf);
        // segment softmax state
        fill_kernel<<<(NNODES + 255) / 256, 256, 0, stream>>>(m, -INFINITY, NNODES);
        hipMemsetAsync(dn, 0, NNODES * sizeof(float), stream);
        // edge phase
        edge_pass1_kernel<<<(NE + 255) / 256, 256, 0, stream>>>(src[et], dst[et], al, ar,
                                                                ebuf, m, NE);
        edge_pass2_kernel<<<(NE + 255) / 256, 256, 0, stream>>>(dst[et], ebuf, m,
                                                                exbuf, dn, NE);
        int agg_grid = (NE + 7) / 8;                   // 8 waves / 256-thread block
        edge_aggregate_kernel<<<agg_grid, 256, 0, stream>>>(src[et], dst[et], exbuf, dn,
                                                            Wh, hout, NE, outf);
    }
}

extern "C" void kernel_launch(void* const* d_in, const int* in_sizes, int n_in,
                              void* d_out, int out_size, void* d_ws, size_t ws_size,
                              hipStream_t stream) {
    const float* x = (const float*)d_in[0];
    const float* W1[2] = {(const float*)d_in[1],  (const float*)d_in[4]};
    const float* b1[2] = {(const float*)d_in[2],  (const float*)d_in[5]};
    const float* a1[2] = {(const float*)d_in[3],  (const float*)d_in[6]};
    const float* W2[2] = {(const float*)d_in[7],  (const float*)d_in[10]};
    const float* b2[2] = {(const float*)d_in[8],  (const float*)d_in[11]};
    const float* a2[2] = {(const float*)d_in[9],  (const float*)d_in[12]};
    const int*  src[2] = {(const int*)d_in[13],   (const int*)d_in[15]};
    const int*  dst[2] = {(const int*)d_in[14],   (const int*)d_in[16]};

    // workspace carve (floats)
    float* ws    = (float*)d_ws;
    float* Wh    = ws;                          // NNODES * 128 (reused both layers)
    float* al    = Wh   + (size_t)NNODES * HIDF;
    float* ar    = al   + NNODES;
    float* m     = ar   + NNODES;
    float* dn    = m    + NNODES;
    float* ebuf  = dn   + NNODES;               // NEDGES
    float* exbuf = ebuf + NEDGES;               // NEDGES
    float* h1    = exbuf + NEDGES;              // NNODES * 128

    // layer 1: IN=128 -> HID=128, then leaky_relu
    run_layer(x, W1, b1, a1, src, dst, h1, INF_F, HIDF,
              Wh, al, ar, m, dn, ebuf, exbuf, stream);
    leaky_relu_kernel<<<((NNODES * HIDF) + 255) / 256, 256, 0, stream>>>(h1, NNODES * HIDF);

    // layer 2: HID=128 -> OUT=64, result straight into d_out
    run_layer(h1, W2, b2, a2, src, dst, (float*)d_out, HIDF, OUTF,
              Wh, al, ar, m, dn, ebuf, exbuf, stream);
}